// GATInfluenceLayer_10780367913777
// MI455X (gfx1250) — compile-verified
//
#include <hip/hip_runtime.h>

#define N_NODES 100000
#define N_EDGES 1600000
#define IN_DIM  128
#define OUT_DIM 64
#define NEG_SLOPE 0.2f

typedef float v2f __attribute__((ext_vector_type(2)));
typedef float v8f __attribute__((ext_vector_type(8)));

// ---- workspace layout (floats) ----
#define WH_OFF   0            // 100000*64 = 6,400,000
#define SSRC_OFF 6400000      // 100,000
#define SDST_OFF 6500000      // 100,000
#define DEN_OFF  6600000      // 100,000
#define EBUF_OFF 6700000      // 1,600,000
#define GMAX_OFF 8300000      // 1 (as unsigned)

// monotone float<->uint ordering for atomicMax on floats of any sign
__device__ __forceinline__ unsigned f2ord(float f) {
    unsigned u = __float_as_uint(f);
    return (u & 0x80000000u) ? ~u : (u | 0x80000000u);
}
__device__ __forceinline__ float ord2f(unsigned u) {
    return (u & 0x80000000u) ? __uint_as_float(~u) : __uint_as_float(u & 0x7fffffffu);
}

// ---------------- init: zero out/denom, reset gmax ----------------
__global__ void k_init(float* __restrict__ out, float* __restrict__ den,
                       unsigned* __restrict__ gmax) {
    int i = blockIdx.x * blockDim.x + threadIdx.x;
    int stride = gridDim.x * blockDim.x;
    for (int j = i; j < N_NODES * OUT_DIM; j += stride) out[j] = 0.0f;
    for (int j = i; j < N_NODES; j += stride) den[j] = 0.0f;
    if (i == 0) *gmax = 0u;   // 0 < ord(-inf) < ord(any finite float)
}

// ---------------- GEMM: Wh = h @ W via V_WMMA_F32_16X16X4_F32 ----------------
// 128 threads = 4 waves; each wave computes a 16x64 tile (4 WMMA accumulators).
// W (128x64 = 32KB) staged in LDS, interleaved so each B fragment is one b64 load:
//   Wl[(k>>1)*128 + 2n + (k&1)] = W[k*64+n]
__global__ void k_gemm(const float* __restrict__ h, const float* __restrict__ Wg,
                       float* __restrict__ Wh) {
    __shared__ float Wl[IN_DIM * OUT_DIM];
    int t = threadIdx.x;
    for (int i = t; i < IN_DIM * OUT_DIM; i += 128) {
        int k = i >> 6, n = i & 63;
        Wl[(k >> 1) * 128 + (n << 1) + (k & 1)] = Wg[i];
    }
    __syncthreads();

    int wave = t >> 5, lane = t & 31;
    int tile = blockIdx.x * 4 + wave;                 // 16-row tile index
    if (tile >= N_NODES / 16) return;

    int mlo = lane & 15;                              // M for A; N for B/C/D
    int khalf = (lane >> 4) << 1;                     // lanes 16-31 handle K+2

    v8f c0 = {}, c1 = {}, c2 = {}, c3 = {};
    const float* hrow = h + (size_t)(tile * 16 + mlo) * IN_DIM;

#pragma unroll 4
    for (int k0 = 0; k0 < IN_DIM; k0 += 4) {
        int kk = k0 + khalf;                          // even
        v2f a = *(const v2f*)(hrow + kk);             // A: M=mlo, K=kk,kk+1
        const float* bbase = &Wl[(kk >> 1) * 128];
        v2f b0 = *(const v2f*)(bbase + ((0 * 16 + mlo) << 1));
        v2f b1 = *(const v2f*)(bbase + ((1 * 16 + mlo) << 1));
        v2f b2 = *(const v2f*)(bbase + ((2 * 16 + mlo) << 1));
        v2f b3 = *(const v2f*)(bbase + ((3 * 16 + mlo) << 1));
        c0 = __builtin_amdgcn_wmma_f32_16x16x4_f32(false, a, false, b0, (short)0, c0, false, false);
        c1 = __builtin_amdgcn_wmma_f32_16x16x4_f32(false, a, false, b1, (short)0, c1, false, false);
        c2 = __builtin_amdgcn_wmma_f32_16x16x4_f32(false, a, false, b2, (short)0, c2, false, false);
        c3 = __builtin_amdgcn_wmma_f32_16x16x4_f32(false, a, false, b3, (short)0, c3, false, false);
    }

    // D layout: VGPR r holds row M = r + 8*(lane>=16), col N = nt*16 + (lane&15)
    union { v8f v; float f[8]; } u;
    int mbase = tile * 16 + 8 * (lane >> 4);
    v8f acc[4] = {c0, c1, c2, c3};
#pragma unroll
    for (int nt = 0; nt < 4; ++nt) {
        u.v = acc[nt];
        int col = nt * 16 + mlo;
#pragma unroll
        for (int r = 0; r < 8; ++r)
            Wh[(size_t)(mbase + r) * OUT_DIM + col] = u.f[r];
    }
}

// ---------------- per-node scores: one wave per node ----------------
__global__ void k_scores(const float* __restrict__ Wh, const float* __restrict__ a,
                         float* __restrict__ ssrc, float* __restrict__ sdst) {
    int lane = threadIdx.x & 31;
    int node = blockIdx.x * 8 + (threadIdx.x >> 5);
    if (node >= N_NODES) return;
    float2 w  = ((const float2*)Wh)[(size_t)node * 32 + lane];
    float2 as = ((const float2*)a)[lane];
    float2 ad = ((const float2*)a)[32 + lane];
    float ss = w.x * as.x + w.y * as.y;
    float sd = w.x * ad.x + w.y * ad.y;
#pragma unroll
    for (int m = 16; m > 0; m >>= 1) {
        ss += __shfl_xor(ss, m, 32);
        sd += __shfl_xor(sd, m, 32);
    }
    if (lane == 0) { ssrc[node] = ss; sdst[node] = sd; }
}

// ---------------- edge pass 1: e = leakyrelu(s_src[row]+s_dst[col]), global max ----
__global__ void k_edge1(const long long* __restrict__ row, const long long* __restrict__ col,
                        const float* __restrict__ ssrc, const float* __restrict__ sdst,
                        float* __restrict__ ebuf, unsigned* __restrict__ gmax) {
    __shared__ float sm[256];
    int i = blockIdx.x * 256 + threadIdx.x;
    long long r = row[i], c = col[i];
    float e = ssrc[r] + sdst[c];
    e = fmaxf(e, 0.0f) + NEG_SLOPE * fminf(e, 0.0f);
    ebuf[i] = e;
    sm[threadIdx.x] = e;
    __syncthreads();
#pragma unroll
    for (int s = 128; s > 0; s >>= 1) {
        if (threadIdx.x < s) sm[threadIdx.x] = fmaxf(sm[threadIdx.x], sm[threadIdx.x + s]);
        __syncthreads();
    }
    if (threadIdx.x == 0) atomicMax(gmax, f2ord(sm[0]));
}

// ---------------- edge pass 2: exp(e - max), segment denom ----------------
__global__ void k_edge2(const long long* __restrict__ row, float* __restrict__ ebuf,
                        float* __restrict__ den, const unsigned* __restrict__ gmax) {
    int i = blockIdx.x * 256 + threadIdx.x;
    float M = ord2f(*gmax);
    float ex = __expf(ebuf[i] - M);
    ebuf[i] = ex;
    atomicAdd(&den[row[i]], ex);
}

// ---------------- edge pass 3: out[row] += alpha * Wh[col] (wave per edge) ------
__global__ void k_edge3(const long long* __restrict__ row, const long long* __restrict__ col,
                        const float* __restrict__ ebuf, const float* __restrict__ den,
                        const float* __restrict__ Wh, float* __restrict__ out) {
    int lane = threadIdx.x & 31;
    int e = blockIdx.x * 8 + (threadIdx.x >> 5);
    if (e >= N_EDGES) return;
    long long r = row[e], c = col[e];
    float alpha = ebuf[e] / (den[r] + 1e-10f);
    float2 w = ((const float2*)Wh)[(size_t)c * 32 + lane];
    float* o = out + (size_t)r * OUT_DIM + (lane << 1);
    atomicAdd(o,     alpha * w.x);
    atomicAdd(o + 1, alpha * w.y);
}

extern "C" void kernel_launch(void* const* d_in, const int* in_sizes, int n_in,
                              void* d_out, int out_size, void* d_ws, size_t ws_size,
                              hipStream_t stream) {
    const float*     h   = (const float*)d_in[0];
    const long long* row = (const long long*)d_in[1];
    const long long* col = (const long long*)d_in[2];
    const float*     W   = (const float*)d_in[3];
    const float*     a   = (const float*)d_in[4];
    float* out = (float*)d_out;
    float* ws  = (float*)d_ws;

    float*    Wh   = ws + WH_OFF;
    float*    ssrc = ws + SSRC_OFF;
    float*    sdst = ws + SDST_OFF;
    float*    den  = ws + DEN_OFF;
    float*    ebuf = ws + EBUF_OFF;
    unsigned* gmax = (unsigned*)(ws + GMAX_OFF);

    k_init  <<<2048, 256, 0, stream>>>(out, den, gmax);
    k_gemm  <<<(N_NODES / 16 + 3) / 4, 128, 0, stream>>>(h, W, Wh);
    k_scores<<<N_NODES / 8, 256, 0, stream>>>(Wh, a, ssrc, sdst);
    k_edge1 <<<N_EDGES / 256, 256, 0, stream>>>(row, col, ssrc, sdst, ebuf, gmax);
    k_edge2 <<<N_EDGES / 256, 256, 0, stream>>>(row, ebuf, den, gmax);
    k_edge3 <<<N_EDGES / 8, 256, 0, stream>>>(row, col, ebuf, den, Wh, out);
}